// MoRStack_76141180223632
// MI455X (gfx1250) — compile-verified
//
#include <hip/hip_runtime.h>
#include <math.h>

// ---------------------------------------------------------------------------
// MoR / ACT stack for MI455X (gfx1250), f32 WMMA (V_WMMA_F32_16X16X4_F32)
// keeps reference f32 numerics on the matrix pipes.
//
// GEMM tiling: BM=64 x BN=64 per block (128 threads = 4 waves, wave32),
// each wave owns a 32x32 quadrant = 2x2 WMMA 16x16 tiles. K staged through
// double-buffered LDS in BK=16 chunks via GLOBAL_LOAD_ASYNC_TO_LDS_B128
// (ASYNCcnt-tracked, VGPR-bypassing) -> one s_wait_asynccnt + one barrier
// per K-stage, with the async fill of stage s+1 overlapping the WMMAs of
// stage s. A-tile LDS stride padded to 20 floats (80B, 16B-aligned) ->
// conflict-free b64 fragment reads across 16 lanes.
// ---------------------------------------------------------------------------

#define BM 64
#define BN 64
#define BK 16
#define BKP 20  // padded LDS row stride for the A tile (floats); 80B, 16B-aligned

typedef __attribute__((ext_vector_type(2))) float v2f;
typedef __attribute__((ext_vector_type(8))) float v8f;

__device__ __forceinline__ float gelu_exact(float x) {
    // matches jax.nn.gelu(approximate=False): 0.5*x*(1+erf(x/sqrt(2)))
    return 0.5f * x * (1.0f + erff(x * 0.70710678118654752f));
}

// Async copy of 16B per lane from global to LDS (gfx1250).
// LDS byte address = low 32 bits of the flat shared-memory pointer
// (ISA 10.2: LDS aperture maps addr[31:0] to the wave's LDS).
__device__ __forceinline__ void async_b128(const void* gptr, void* lptr) {
    const unsigned long long ga = (unsigned long long)(uintptr_t)gptr;
    const unsigned int la = (unsigned int)(uintptr_t)lptr;
    asm volatile("global_load_async_to_lds_b128 %0, %1, off"
                 :: "v"(la), "v"(ga)
                 : "memory");
}

__device__ __forceinline__ void wait_async0() {
    asm volatile("s_wait_asynccnt 0x0" ::: "memory");
}

// MODE 0: Out[m,n] = gelu(acc + bias[n])            (FFN1 / router hidden)
// MODE 1: xn = acc + bias[n];
//         wtok ? Out[m,n] = w[m]*xn + (1-w[m])*Out[m,n] : Out[m,n] = xn
template <int MODE>
__global__ __launch_bounds__(128) void gemm_wmma(
    const float* __restrict__ A,     // M x K, row-major (row stride K)
    const float* __restrict__ W,     // K x N, row-major (row stride N)
    const float* __restrict__ bias,  // N
    float* __restrict__ Out,         // M x N
    const float* __restrict__ wtok,  // per-token blend weight (MODE 1), or null
    int M, int N, int K)
{
    __shared__ __align__(16) float As[2][BM * BKP];  // 2 x (64 x 16, pad 20)
    __shared__ __align__(16) float Bs[2][BK * BN];   // 2 x (16 x 64)

    const int tid  = threadIdx.x;
    const int lane = tid & 31;
    const int wid  = tid >> 5;
    const int lr   = lane & 15;  // row/col within 16-wide stripe
    const int hi   = lane >> 4;  // half-wave selector (K pairs / M+8)
    const int wm   = wid & 1;    // wave row quadrant
    const int wn   = wid >> 1;   // wave col quadrant

    const int m0 = blockIdx.y * BM;
    const int n0 = blockIdx.x * BN;

    v8f c[2][2];
    const v8f vzero = {0.f, 0.f, 0.f, 0.f, 0.f, 0.f, 0.f, 0.f};
#pragma unroll
    for (int mt = 0; mt < 2; ++mt)
#pragma unroll
        for (int nt = 0; nt < 2; ++nt) c[mt][nt] = vzero;

    // Issue the async fill for one K-stage: A tile 64x16 + B tile 16x64,
    // 4 async b128 per thread (16B/lane), ASYNCcnt <= 4 per wave.
    auto issue = [&](int kb, int buf) {
#pragma unroll
        for (int it = 0; it < 2; ++it) {  // A tile: 256 x b128 lanes
            const int idx = tid + it * 128;
            const int row = idx >> 2;
            const int c4  = idx & 3;
            async_b128(A + (size_t)(m0 + row) * K + kb + c4 * 4,
                       &As[buf][row * BKP + c4 * 4]);
        }
#pragma unroll
        for (int it = 0; it < 2; ++it) {  // B tile: 256 x b128 lanes
            const int idx = tid + it * 128;
            const int row = idx >> 4;
            const int c4  = idx & 15;
            async_b128(W + (size_t)(kb + row) * N + n0 + c4 * 4,
                       &Bs[buf][row * BN + c4 * 4]);
        }
    };

    const int nstages = K / BK;
    issue(0, 0);

    for (int s = 0; s < nstages; ++s) {
        // Own async fills done -> barrier makes every wave's fills visible.
        wait_async0();
        __syncthreads();
        // Overlap: fill the other buffer while this one feeds the WMMAs.
        // Safe: every wave finished reading buf[(s+1)&1] in stage s-1 before
        // reaching the barrier above.
        if (s + 1 < nstages) issue((s + 1) * BK, (s + 1) & 1);

        const float* as = As[s & 1];
        const float* bs = Bs[s & 1];
#pragma unroll
        for (int kk = 0; kk < 4; ++kk) {
            const int k0 = kk * 4 + 2 * hi;  // lanes 0-15: K{0,1}; 16-31: K{2,3}
            v2f a[2], b[2];
#pragma unroll
            for (int mt = 0; mt < 2; ++mt) {
                const int row = wm * 32 + mt * 16 + lr;
                a[mt] = *(const v2f*)&as[row * BKP + k0];
            }
#pragma unroll
            for (int nt = 0; nt < 2; ++nt) {
                const int col = wn * 32 + nt * 16 + lr;
                b[nt].x = bs[k0 * BN + col];
                b[nt].y = bs[(k0 + 1) * BN + col];
            }
#pragma unroll
            for (int mt = 0; mt < 2; ++mt)
#pragma unroll
                for (int nt = 0; nt < 2; ++nt)
                    c[mt][nt] = __builtin_amdgcn_wmma_f32_16x16x4_f32(
                        false, a[mt], false, b[nt], (short)0, c[mt][nt],
                        false, false);
        }
    }

    // --- epilogue: C/D layout: element r of v8f -> row (r + 8*hi), col lr ---
#pragma unroll
    for (int mt = 0; mt < 2; ++mt) {
#pragma unroll
        for (int nt = 0; nt < 2; ++nt) {
            const int col = n0 + wn * 32 + nt * 16 + lr;
            const float bv = bias[col];
#pragma unroll
            for (int r = 0; r < 8; ++r) {
                const int row = m0 + wm * 32 + mt * 16 + r + 8 * hi;
                const size_t o = (size_t)row * N + col;
                const float acc = c[mt][nt][r] + bv;
                if (MODE == 0) {
                    Out[o] = gelu_exact(acc);
                } else {
                    if (wtok) {
                        const float wv = wtok[row];
                        Out[o] = wv * acc + (1.0f - wv) * Out[o];
                    } else {
                        Out[o] = acc;
                    }
                }
            }
        }
    }
}

// Router head + ACT state update: one wave32 per token.
__global__ __launch_bounds__(256) void router_act_kernel(
    const float* __restrict__ Hr,    // M x 512 (gelu'd router hidden)
    const float* __restrict__ Rw2,   // 512
    const float* __restrict__ Rb2,   // 1
    float* __restrict__ wtok, float* __restrict__ wsum,
    float* __restrict__ cum, float* __restrict__ still, int M)
{
    const int wid  = threadIdx.x >> 5;
    const int lane = threadIdx.x & 31;
    const int t = blockIdx.x * 8 + wid;
    if (t >= M) return;

    const float* h = Hr + (size_t)t * 512;
    float acc = 0.0f;
#pragma unroll 4
    for (int m = lane; m < 512; m += 32) acc += h[m] * Rw2[m];
#pragma unroll
    for (int off = 16; off > 0; off >>= 1) acc += __shfl_down(acc, off, 32);

    if (lane == 0) {
        const float logit = acc + Rb2[0];
        const float p = 1.0f / (1.0f + expf(-logit));
        const float cv = cum[t];
        const float sv = still[t];
        const float rem = fmaxf(1.0f - cv, 0.0f);
        const bool halt = (cv + p) >= 0.99f;              // ACT_TAU
        const float wv = (halt ? rem : p) * sv;
        wtok[t] = wv;
        wsum[t] += wv;
        cum[t]  = cv + wv;                                 // ref adds weight
        still[t] = halt ? 0.0f : sv;
    }
}

__global__ void init_state(float* __restrict__ wsum, float* __restrict__ cum,
                           float* __restrict__ still, int M)
{
    const int t = blockIdx.x * blockDim.x + threadIdx.x;
    if (t < M) {
        wsum[t] = 1.0f;  // mandatory first pass counted
        cum[t]  = 0.0f;
        still[t] = 1.0f;
    }
}

__global__ __launch_bounds__(256) void finalize_kernel(
    const float* __restrict__ wsum, float* __restrict__ out, int M)
{
    __shared__ float red[256];
    float s = 0.0f;
    for (int i = threadIdx.x; i < M; i += 256) s += wsum[i];
    red[threadIdx.x] = s;
    __syncthreads();
#pragma unroll
    for (int st = 128; st > 0; st >>= 1) {
        if (threadIdx.x < st) red[threadIdx.x] += red[threadIdx.x + st];
        __syncthreads();
    }
    if (threadIdx.x == 0) {
        const float avg = red[0] / (float)M;
        const float d = avg - 2.5f;                        // TARGET_DEPTH
        const float r = d > 0.0f ? d : 0.0f;
        out[0] = 0.01f * r * r;                            // BUDGET_WEIGHT
    }
}

extern "C" void kernel_launch(void* const* d_in, const int* in_sizes, int n_in,
                              void* d_out, int out_size, void* d_ws, size_t ws_size,
                              hipStream_t stream)
{
    (void)in_sizes; (void)n_in; (void)out_size; (void)ws_size;
    const float* x_in = (const float*)d_in[0];  // (4,2048,1024)
    const float* Wb1  = (const float*)d_in[1];  // (1024,4096)
    const float* bb1  = (const float*)d_in[2];  // (4096,)
    const float* Wb2  = (const float*)d_in[3];  // (4096,1024)
    const float* bb2  = (const float*)d_in[4];  // (1024,)
    const float* Rw1  = (const float*)d_in[5];  // (3,1025,512)
    const float* Rb1  = (const float*)d_in[6];  // (3,512)
    const float* Rw2  = (const float*)d_in[7];  // (3,512,1)
    const float* Rb2  = (const float*)d_in[8];  // (3,1)

    const int H = 1024, DFF = 4096, HR = 512;
    const int M = 4 * 2048;  // 8192 tokens

    float* x = (float*)d_out;                 // final x lives in d_out
    float* budget = x + (size_t)M * H;        // scalar at tail

    // workspace layout (floats): Y(128MB) | Hr(16MB) | wtok | wsum | cum | still
    float* ws   = (float*)d_ws;
    float* Y    = ws;  ws += (size_t)M * DFF;
    float* Hr   = ws;  ws += (size_t)M * HR;
    float* wtok = ws;  ws += M;
    float* wsum = ws;  ws += M;
    float* cum  = ws;  ws += M;
    float* still = ws; ws += M;

    const dim3 blk(128);
    const dim3 g1(DFF / BN, M / BM);  // 64 x 128 blocks (FFN1)
    const dim3 g2(H / BN, M / BM);    // 16 x 128 blocks (FFN2)
    const dim3 gr(HR / BN, M / BM);   //  8 x 128 blocks (router hidden)

    // mandatory first body pass: x = body(x_in)
    gemm_wmma<0><<<g1, blk, 0, stream>>>(x_in, Wb1, bb1, Y, nullptr, M, DFF, H);
    gemm_wmma<1><<<g2, blk, 0, stream>>>(Y, Wb2, bb2, x, nullptr, M, H, DFF);
    init_state<<<(M + 255) / 256, 256, 0, stream>>>(wsum, cum, still, M);

    for (int i = 0; i < 3; ++i) {
        // concat'd feature is zero -> Rw1 row 1024 contributes nothing; use
        // rows 0..1023 directly (row stride = 512).
        const float* rw1 = Rw1 + (size_t)i * 1025 * 512;
        const float* rb1 = Rb1 + (size_t)i * 512;
        const float* rw2 = Rw2 + (size_t)i * 512;
        const float* rb2 = Rb2 + i;

        gemm_wmma<0><<<gr, blk, 0, stream>>>(x, rw1, rb1, Hr, nullptr, M, HR, H);
        router_act_kernel<<<M / 8, 256, 0, stream>>>(Hr, rw2, rb2, wtok, wsum,
                                                     cum, still, M);
        gemm_wmma<0><<<g1, blk, 0, stream>>>(x, Wb1, bb1, Y, nullptr, M, DFF, H);
        gemm_wmma<1><<<g2, blk, 0, stream>>>(Y, Wb2, bb2, x, wtok, M, H, DFF);
    }

    finalize_kernel<<<1, 256, 0, stream>>>(wsum, budget, M);
}